// BackboneModule_5317169512894
// MI455X (gfx1250) — compile-verified
//
#include <hip/hip_runtime.h>
#include <cstdint>

// BackboneModule: per-residue rigid-frame chain + atom placement.
// Memory-bound (AI ~3.5 FLOP/B): optimized for the MI455X data-movement path
// (async global->LDS staging, LDS frame storage for dynamic tree indexing,
// coalesced b128 async writeback). WMMA is intentionally not used: 3x3 frame
// algebra with per-lane dynamic indices cannot utilize 16x16 tiles.
//
// Each block handles RPT slices of BLOCK residues; the 16.8 KB of per-type
// tables are staged once per block (amortized ~65 B/residue of L2 traffic).

#define BLOCK    64
#define RPT      4
#define NRES     21
#define MAXR     8
#define MAXA     24
#define FRSTRIDE 100   // dwords per thread in frame buffer (16B aligned, 96 used)
#define ATSTRIDE 76    // floats per thread in atom staging (16B aligned, 72 used)

#define AS_GLOBAL __attribute__((address_space(1)))
#define AS_LDS    __attribute__((address_space(3)))

typedef int v4i __attribute__((vector_size(16)));

struct Frame { float r[3][3]; float t[3]; };

struct SMem {
  float tf[NRES * MAXR * 12];   // transforms_tensor  (8064 B)
  float rg[NRES * MAXA * 3];    // rigids_tensor      (6048 B)
  int   td[NRES * MAXR];        // transforms_dep     ( 672 B)
  int   rd[NRES * MAXA];        // rigids_dep         (2016 B)
  float fr[BLOCK * FRSTRIDE];   // per-thread frames  (25600 B)
  union {                       // time-shared per slice: inputs then atom staging
    struct { float bb[BLOCK * 12]; float sc[BLOCK * 14]; float p0[BLOCK * 3]; } in;
    float oat[BLOCK * ATSTRIDE];                        // (19456 B)
  } dyn;
};                              // total 61856 B

static __device__ inline void wait_asynccnt0() {
#if __has_builtin(__builtin_amdgcn_s_wait_asynccnt)
  __builtin_amdgcn_s_wait_asynccnt(0);
#else
  asm volatile("s_wait_asynccnt 0" ::: "memory");
#endif
}

// Cooperative async copy global -> LDS in 16B chunks (nbytes % 16 == 0).
static __device__ inline void async_copy_g2l(void* lds_dst, const void* gsrc,
                                             int nbytes, int tid, int nthr) {
#if __has_builtin(__builtin_amdgcn_global_load_async_to_lds_b128)
  for (int o = tid * 16; o < nbytes; o += nthr * 16) {
    __builtin_amdgcn_global_load_async_to_lds_b128(
        (AS_GLOBAL v4i*)(uintptr_t)((const char*)gsrc + o),
        (AS_LDS v4i*)((char*)lds_dst + o), 0, 0);
  }
#else
  unsigned lbase = (unsigned)(uintptr_t)(AS_LDS char*)lds_dst;
  for (int o = tid * 16; o < nbytes; o += nthr * 16) {
    unsigned lo = lbase + (unsigned)o;
    unsigned long long ga = (unsigned long long)(uintptr_t)((const char*)gsrc + o);
    asm volatile("global_load_async_to_lds_b128 %0, %1, off"
                 :: "v"(lo), "v"(ga) : "memory");
  }
#endif
}

static __device__ inline void async_store_l2g_b128(void* gdst, const void* lds_src) {
#if __has_builtin(__builtin_amdgcn_global_store_async_from_lds_b128)
  __builtin_amdgcn_global_store_async_from_lds_b128(
      (AS_GLOBAL v4i*)(uintptr_t)gdst,
      (AS_LDS v4i*)(char*)lds_src, 0, 0);
#else
  unsigned lo = (unsigned)(uintptr_t)(AS_LDS char*)(char*)lds_src;
  unsigned long long ga = (unsigned long long)(uintptr_t)gdst;
  asm volatile("global_store_async_from_lds_b128 %0, %1, off"
               :: "v"(ga), "v"(lo) : "memory");
#endif
}

static __device__ inline Frame ld_frame(const float* p) {  // p 16B aligned (LDS)
  const float4* q = (const float4*)p;
  float4 a = q[0], b = q[1], c = q[2];
  Frame f;
  f.r[0][0] = a.x; f.r[0][1] = a.y; f.r[0][2] = a.z;
  f.r[1][0] = a.w; f.r[1][1] = b.x; f.r[1][2] = b.y;
  f.r[2][0] = b.z; f.r[2][1] = b.w; f.r[2][2] = c.x;
  f.t[0] = c.y; f.t[1] = c.z; f.t[2] = c.w;
  return f;
}

static __device__ inline void st_frame(float* p, const Frame& f) {
  float4* q = (float4*)p;
  q[0] = make_float4(f.r[0][0], f.r[0][1], f.r[0][2], f.r[1][0]);
  q[1] = make_float4(f.r[1][1], f.r[1][2], f.r[2][0], f.r[2][1]);
  q[2] = make_float4(f.r[2][2], f.t[0],    f.t[1],    f.t[2]);
}

// D = X o Y:  D.R = X.R*Y.R ; D.t = X.R*Y.t + X.t
static __device__ inline Frame combine(const Frame& X, const Frame& Y) {
  Frame D;
#pragma unroll
  for (int i = 0; i < 3; ++i) {
#pragma unroll
    for (int j = 0; j < 3; ++j)
      D.r[i][j] = X.r[i][0]*Y.r[0][j] + X.r[i][1]*Y.r[1][j] + X.r[i][2]*Y.r[2][j];
    D.t[i] = X.r[i][0]*Y.t[0] + X.r[i][1]*Y.t[1] + X.r[i][2]*Y.t[2] + X.t[i];
  }
  return D;
}

__global__ __launch_bounds__(BLOCK) void backbone_kernel(
    const int*   __restrict__ rtype,
    const float* __restrict__ bb,
    const float* __restrict__ pos0,
    const float* __restrict__ sc,
    const float* __restrict__ tft,
    const float* __restrict__ rgt,
    const int*   __restrict__ tdep,
    const int*   __restrict__ rdep,
    float*       __restrict__ out_atoms,
    float*       __restrict__ out_frames)
{
  __shared__ __align__(16) SMem sm;
  const int t   = threadIdx.x;
  const int blk = blockIdx.x;

  // ---- Stage the per-type tables once per block (async global->LDS) ----
  async_copy_g2l(sm.tf, tft, (int)sizeof(sm.tf), t, BLOCK);
  async_copy_g2l(sm.rg, rgt, (int)sizeof(sm.rg), t, BLOCK);
  async_copy_g2l(sm.td, tdep, (int)sizeof(sm.td), t, BLOCK);
  async_copy_g2l(sm.rd, rdep, (int)sizeof(sm.rd), t, BLOCK);

#pragma unroll 1
  for (int rep = 0; rep < RPT; ++rep) {
    const int slice = blk * RPT + rep;        // 64-residue slice index
    const int n     = slice * BLOCK + t;      // this thread's residue

    // ---- Stage this slice's bb / sc / pos0 via async global->LDS ----
    async_copy_g2l(sm.dyn.in.bb, bb   + (size_t)slice * (BLOCK * 12), BLOCK * 12 * 4, t, BLOCK);
    async_copy_g2l(sm.dyn.in.sc, sc   + (size_t)slice * (BLOCK * 14), BLOCK * 14 * 4, t, BLOCK);
    async_copy_g2l(sm.dyn.in.p0, pos0 + (size_t)slice * (BLOCK * 3),  BLOCK * 3 * 4,  t, BLOCK);
    wait_asynccnt0();          // this wave's async loads (incl. tables on rep 0) done
    __syncthreads();           // ... and everyone else's

    const int rt = rtype[n];   // coalesced b32
    const float p0x = sm.dyn.in.p0[t * 3 + 0];
    const float p0y = sm.dyn.in.p0[t * 3 + 1];
    const float p0z = sm.dyn.in.p0[t * 3 + 2];

    float4 b0, b1, b2;
    {
      const float4* pb = (const float4*)&sm.dyn.in.bb[t * 12];
      b0 = pb[0]; b1 = pb[1]; b2 = pb[2];
    }
    float cs[14];
    {
      const float2* ps = (const float2*)&sm.dyn.in.sc[t * 14];
#pragma unroll
      for (int i = 0; i < 7; ++i) { float2 v = ps[i]; cs[2*i] = v.x; cs[2*i+1] = v.y; }
    }

    float*       fb  = &sm.fr[t * FRSTRIDE];
    const float* Tb  = &sm.tf[rt * (MAXR * 12)];
    const int*   tdl = &sm.td[rt * MAXR];

    // ---- Frame 0: combine(T0, [bb_R ; bb_t + pos0]) ----
    {
      Frame T0 = ld_frame(Tb);
      Frame o0;
      o0.r[0][0] = b0.x; o0.r[0][1] = b0.y; o0.r[0][2] = b0.z;
      o0.r[1][0] = b0.w; o0.r[1][1] = b1.x; o0.r[1][2] = b1.y;
      o0.r[2][0] = b1.z; o0.r[2][1] = b1.w; o0.r[2][2] = b2.x;
      o0.t[0] = b2.y + p0x; o0.t[1] = b2.z + p0y; o0.t[2] = b2.w + p0z;
      Frame f0 = combine(T0, o0);
      st_frame(fb, f0);
      // second output: opr[:,0] -> direct 3x b128 stores (48 B/residue)
      float4* po = (float4*)(out_frames + (size_t)n * 12);
      po[0] = make_float4(f0.r[0][0], f0.r[0][1], f0.r[0][2], f0.r[1][0]);
      po[1] = make_float4(f0.r[1][1], f0.r[1][2], f0.r[2][0], f0.r[2][1]);
      po[2] = make_float4(f0.r[2][2], f0.t[0],    f0.t[1],    f0.t[2]);
    }

    // ---- Frames 1..7: torsion pre-multiply (rot-x column trick) + chain ----
#pragma unroll
    for (int k = 1; k < MAXR; ++k) {
      Frame Tk = ld_frame(Tb + k * 12);
      const float c = cs[2*(k-1)], s = cs[2*(k-1)+1];
      Frame cur;
#pragma unroll
      for (int i = 0; i < 3; ++i) {
        cur.r[i][0] = Tk.r[i][0];
        cur.r[i][1] = c * Tk.r[i][1] + s * Tk.r[i][2];
        cur.r[i][2] = c * Tk.r[i][2] - s * Tk.r[i][1];
        cur.t[i]    = Tk.t[i];
      }
      const int dep = tdl[k];            // dep < k: already finalized
      Frame prev = ld_frame(fb + dep * 12);
      Frame nf = combine(prev, cur);
      st_frame(fb + k * 12, nf);
    }

    __syncthreads();   // all dyn.in reads complete -> reuse union as atom staging

    // ---- Atoms: p = R[rdep]*v + t[rdep], staged to LDS ----
    const int*   rdl = &sm.rd[rt * MAXA];
    const float* rgl = &sm.rg[rt * (MAXA * 3)];
    float*       oat = &sm.dyn.oat[t * ATSTRIDE];
#pragma unroll
    for (int a = 0; a < MAXA; ++a) {
      const int rid = rdl[a];
      Frame F = ld_frame(fb + rid * 12);
      const float vx = rgl[a*3+0], vy = rgl[a*3+1], vz = rgl[a*3+2];
      oat[a*3+0] = F.r[0][0]*vx + F.r[0][1]*vy + F.r[0][2]*vz + F.t[0];
      oat[a*3+1] = F.r[1][0]*vx + F.r[1][1]*vy + F.r[1][2]*vz + F.t[1];
      oat[a*3+2] = F.r[2][0]*vx + F.r[2][1]*vy + F.r[2][2]*vz + F.t[2];
    }

    __syncthreads();

    // ---- Coalesced writeback: 18 b128 chunks per residue, lane-contiguous ----
    float* abase = out_atoms + (size_t)slice * (BLOCK * 72);
#pragma unroll
    for (int i = 0; i < 18; ++i) {
      const int q  = i * BLOCK + t;       // float4 chunk index within slice region
      const int tt = q / 18;              // source residue (thread) in this slice
      const int r  = (q % 18) * 4;        // float offset within residue
      async_store_l2g_b128(abase + (size_t)q * 4, &sm.dyn.oat[tt * ATSTRIDE + r]);
    }
    wait_asynccnt0();   // stores have consumed dyn.oat
    __syncthreads();    // safe to overwrite union with next slice's inputs
  }
}

extern "C" void kernel_launch(void* const* d_in, const int* in_sizes, int n_in,
                              void* d_out, int out_size, void* d_ws, size_t ws_size,
                              hipStream_t stream) {
  const int*   rtype = (const int*)d_in[0];
  const float* bb    = (const float*)d_in[1];
  const float* pos0  = (const float*)d_in[2];
  const float* sc    = (const float*)d_in[3];
  const float* tft   = (const float*)d_in[4];
  const float* rgt   = (const float*)d_in[5];
  const int*   tdep  = (const int*)d_in[6];
  const int*   rdep  = (const int*)d_in[7];

  const int N = in_sizes[0];            // 524288, divisible by BLOCK*RPT
  float* out_atoms  = (float*)d_out;                    // (N, 24, 3)
  float* out_frames = out_atoms + (size_t)N * 72;       // (N, 4, 3)

  dim3 grid(N / (BLOCK * RPT)), block(BLOCK);
  hipLaunchKernelGGL(backbone_kernel, grid, block, 0, stream,
                     rtype, bb, pos0, sc, tft, rgt, tdep, rdep,
                     out_atoms, out_frames);
}